// SinkhornLoss_67001489817679
// MI455X (gfx1250) — compile-verified
//
#include <hip/hip_runtime.h>
#include <hip/hip_bf16.h>

// ---------------------------------------------------------------------------
// Sinkhorn divergence loss, MI455X (gfx1250, wave32).
// Dominant work: row-wise logsumexp over the NxN Gram-derived cost matrix.
// Gram tiles via V_WMMA_F32_16X16X4_F32; logsumexp kept in exp2 domain with a
// per-chunk two-pass (max, then sum) so each element costs ONE v_exp_f32.
// Per-row constant terms are hoisted out of the inner loops (shift-invariance
// of logsumexp) and applied once in the epilogue.
// ---------------------------------------------------------------------------

constexpr int kB = 2;
constexpr int kN = 4096;
constexpr int kV = 32768;
constexpr int kK = 128;
constexpr int kBC = 6;           // (b, c) pairs, bc = b*3 + c

constexpr float kLOG2E = 1.4426950408889634f;   // log2(e)
constexpr float kLN2   = 0.6931471805599453f;   // ln(2)

// device exp2 / log2 mapping straight to v_exp_f32 / v_log_f32
__device__ __forceinline__ float dexp2(float x) { return __builtin_amdgcn_exp2f(x); }
__device__ __forceinline__ float dlog2(float x) { return __builtin_amdgcn_logf(x); }

// ---- workspace layout (float offsets into d_ws) ---------------------------
constexpr int OFF_SCRATCH = 0;                      // 6*V   topk mutable copy
constexpr int OFF_VALS    = OFF_SCRATCH + kBC*kV;   // 6*K   topk values (sorted desc)
constexpr int OFF_PTS4    = OFF_VALS + kBC*kK;      // 6*K*4 (p0,p1,p2,|p|^2), 16B aligned
constexpr int OFF_BLOG    = OFF_PTS4 + kBC*kK*4;    // 6*K   log(max(vals,1e-20))
constexpr int OFF_ALOG    = OFF_BLOG + kBC*kK;      // 6*N   log(max(w,1e-20))
constexpr int OFF_F       = OFF_ALOG + kBC*kN;      // 6*N   f (loop_ab)
constexpr int OFF_FTMP    = OFF_F    + kBC*kN;      // 6*N   ft scratch
constexpr int OFF_G       = OFF_FTMP + kBC*kN;      // 6*K   g (loop_ab)
constexpr int OFF_GTMP    = OFF_G    + kBC*kK;      // 6*K   gt scratch
constexpr int OFF_FA0A    = OFF_GTMP + kBC*kK;      // 6*N   loop_sym ping
constexpr int OFF_FA0B    = OFF_FA0A + kBC*kN;      // 6*N   loop_sym pong
constexpr int OFF_FXAB    = OFF_FA0B + kBC*kN;      // 6*N   F_ab
constexpr int OFF_GXAB    = OFF_FXAB + kBC*kN;      // 6*K   G_ab
constexpr int OFF_FXA     = OFF_GXAB + kBC*kK;      // 6*N   F_a
constexpr int OFF_GXB     = OFF_FXA  + kBC*kN;      // 6*K   G_b
constexpr int OFF_PART    = OFF_GXB  + kBC*kK;      // 6     per-(b,c) partial sums

typedef float v2f __attribute__((ext_vector_type(2)));
typedef float v8f __attribute__((ext_vector_type(8)));

__constant__ float d_EPSS[12] = {4.0f, 1.0f, 0.25f, 0.0625f, 0.015625f, 0.00390625f,
                                 0.0025f, 0.0025f, 0.0025f, 0.0025f, 0.0025f, 0.0025f};

// ---------------------------------------------------------------------------
// 1) top-128 per (b,c) over V=32768 by iterative block argmax (sorted output,
//    ties -> lowest index, matching jax.lax.top_k).
// ---------------------------------------------------------------------------
__global__ void topk_kernel(const float* __restrict__ yhat,
                            const float* __restrict__ vox,
                            float* __restrict__ ws) {
  const int bc = blockIdx.x, b = bc / 3, c = bc % 3;
  float* scratch = ws + OFF_SCRATCH + (size_t)bc * kV;
  const float* src = yhat + ((size_t)b * 3 + c) * kV;
  for (int v = threadIdx.x; v < kV; v += 256) scratch[v] = src[v];
  __syncthreads();

  __shared__ float sv[256];
  __shared__ int   si[256];
  float*  vals = ws + OFF_VALS + bc * kK;
  float*  blog = ws + OFF_BLOG + bc * kK;
  float4* pts4 = ((float4*)(ws + OFF_PTS4)) + bc * kK;

  for (int k = 0; k < kK; ++k) {
    float bv = -3.4e38f; int bi = 0x7fffffff;
    for (int v = threadIdx.x; v < kV; v += 256) {
      float x = scratch[v];
      if (x > bv) { bv = x; bi = v; }   // ascending scan keeps lowest index on tie
    }
    sv[threadIdx.x] = bv; si[threadIdx.x] = bi;
    __syncthreads();
    for (int s = 128; s > 0; s >>= 1) {
      if (threadIdx.x < s) {
        float ov = sv[threadIdx.x + s]; int oi = si[threadIdx.x + s];
        float mv = sv[threadIdx.x];     int mi = si[threadIdx.x];
        if (ov > mv || (ov == mv && oi < mi)) { sv[threadIdx.x] = ov; si[threadIdx.x] = oi; }
      }
      __syncthreads();
    }
    if (threadIdx.x == 0) {
      float val = sv[0]; int idx = si[0];
      vals[k] = val;
      blog[k] = logf(fmaxf(val, 1e-20f));
      float p0 = vox[idx * 3 + 0], p1 = vox[idx * 3 + 1], p2 = vox[idx * 3 + 2];
      pts4[k] = make_float4(p0, p1, p2, p0 * p0 + p1 * p1 + p2 * p2);
      scratch[idx] = -3.4e38f;
    }
    __syncthreads();
  }
}

// ---------------------------------------------------------------------------
// 2) a_log per (b,c); zero-init f, fa0.  g zeroed for j<K.
// ---------------------------------------------------------------------------
__global__ void prep_kernel(const float* __restrict__ y, float* __restrict__ ws) {
  const int bc = blockIdx.y, b = bc / 3, c = bc % 3;
  const int i = blockIdx.x * 256 + threadIdx.x;
  float w = y[((size_t)b * kN + i) * 6 + 3 + c];
  ws[OFF_ALOG + bc * kN + i] = logf(fmaxf(w, 1e-20f));
  ws[OFF_F    + bc * kN + i] = 0.0f;
  ws[OFF_FA0A + bc * kN + i] = 0.0f;
  if (i < kK) ws[OFF_G + bc * kK + i] = 0.0f;
}

// ---------------------------------------------------------------------------
// 3) softmin over C_xy rows: ft[i] = -eps*LSE_j( b_log[j]+g[j]/eps - C_xy[i,j]/eps )
//    exp2 domain; row constant hhi applied only in the epilogue.
// ---------------------------------------------------------------------------
__global__ void ftxy_kernel(const float* __restrict__ y, float* __restrict__ ws,
                            int out_off, float eps) {
  const int bc = blockIdx.y, b = bc / 3;
  const float inv = 1.0f / eps;
  const float s1  = inv * kLOG2E;
  __shared__ float4 p4[kK];
  __shared__ float  hh[kK];                 // (h_j - 0.5|p_j|^2/eps) * log2e
  if (threadIdx.x < kK) {
    float4 p = ((const float4*)(ws + OFF_PTS4))[bc * kK + threadIdx.x];
    p4[threadIdx.x] = p;
    float h = ws[OFF_BLOG + bc * kK + threadIdx.x] + ws[OFF_G + bc * kK + threadIdx.x] * inv;
    hh[threadIdx.x] = (h - 0.5f * p.w * inv) * kLOG2E;
  }
  __syncthreads();
  const int i = blockIdx.x * 256 + threadIdx.x;
  const float* yp = y + ((size_t)b * kN + i) * 6;
  float x0 = yp[0], x1 = yp[1], x2 = yp[2];
  float hhi = -0.5f * (x0 * x0 + x1 * x1 + x2 * x2) * s1;   // row constant
  float m = -3.0e38f;
  for (int j = 0; j < kK; ++j) {
    float4 p = p4[j];
    float u = fmaf(x0 * p.x + x1 * p.y + x2 * p.z, s1, hh[j]);
    m = fmaxf(m, u);
  }
  float s = 0.0f;
  for (int j = 0; j < kK; ++j) {
    float4 p = p4[j];
    float u = fmaf(x0 * p.x + x1 * p.y + x2 * p.z, s1, hh[j]);
    s += dexp2(u - m);
  }
  ws[out_off + bc * kN + i] = -eps * kLN2 * (m + hhi + dlog2(s));
}

// ---------------------------------------------------------------------------
// 4) softmin over C_xy cols: gt[j] = -eps*LSE_i( a_log[i]+f[i]/eps - C_xy[i,j]/eps )
//    one block per (b,c,j), block tree reduction (deterministic).
// ---------------------------------------------------------------------------
__global__ void gtxy_kernel(const float* __restrict__ y, float* __restrict__ ws,
                            int out_off, float eps) {
  const int bc = blockIdx.y, b = bc / 3;
  const int j = blockIdx.x;
  const float inv = 1.0f / eps;
  const float s1  = inv * kLOG2E;
  float4 p = ((const float4*)(ws + OFF_PTS4))[bc * kK + j];
  __shared__ float red[256];

  float m = -3.0e38f;
  for (int i = threadIdx.x; i < kN; i += 256) {
    const float* yp = y + ((size_t)b * kN + i) * 6;
    float x0 = yp[0], x1 = yp[1], x2 = yp[2];
    float xn = x0 * x0 + x1 * x1 + x2 * x2;
    float h2 = ws[OFF_ALOG + bc * kN + i] + ws[OFF_F + bc * kN + i] * inv;
    float C  = 0.5f * (xn + p.w) - (x0 * p.x + x1 * p.y + x2 * p.z);
    float u  = (h2 - C * inv) * kLOG2E;
    m = fmaxf(m, u);
  }
  red[threadIdx.x] = m; __syncthreads();
  for (int s = 128; s > 0; s >>= 1) {
    if (threadIdx.x < s) red[threadIdx.x] = fmaxf(red[threadIdx.x], red[threadIdx.x + s]);
    __syncthreads();
  }
  m = red[0]; __syncthreads();

  float sum = 0.0f;
  for (int i = threadIdx.x; i < kN; i += 256) {
    const float* yp = y + ((size_t)b * kN + i) * 6;
    float x0 = yp[0], x1 = yp[1], x2 = yp[2];
    float xn = x0 * x0 + x1 * x1 + x2 * x2;
    float h2 = ws[OFF_ALOG + bc * kN + i] + ws[OFF_F + bc * kN + i] * inv;
    float C  = 0.5f * (xn + p.w) - (x0 * p.x + x1 * p.y + x2 * p.z);
    float u  = (h2 - C * inv) * kLOG2E;
    sum += dexp2(u - m);
  }
  red[threadIdx.x] = sum; __syncthreads();
  for (int s = 128; s > 0; s >>= 1) {
    if (threadIdx.x < s) red[threadIdx.x] += red[threadIdx.x + s];
    __syncthreads();
  }
  if (threadIdx.x == 0) ws[out_off + bc * kK + j] = -eps * kLN2 * (m + dlog2(red[0]));
}

// 5) f = 0.5*(f+ft); g = 0.5*(g+gt)
__global__ void upd_kernel(float* __restrict__ ws) {
  const int bc = blockIdx.y;
  const int i = blockIdx.x * 256 + threadIdx.x;
  ws[OFF_F + bc * kN + i] = 0.5f * (ws[OFF_F + bc * kN + i] + ws[OFF_FTMP + bc * kN + i]);
  if (i < kK)
    ws[OFF_G + bc * kK + i] = 0.5f * (ws[OFF_G + bc * kK + i] + ws[OFF_GTMP + bc * kK + i]);
}

// ---------------------------------------------------------------------------
// 6) HEAVY KERNEL: softmin over C_xx (NxN) with WMMA Gram tiles.
//    Block = 256 threads (8 waves) = 128 rows; wave owns 16 rows.
//    Inner-loop value (row constant hhi hoisted to epilogue):
//      u_ij = (x_i.x_j)*(log2e/eps) + (h_j - 0.5|x_j|^2/eps)*log2e
//    Per 1024-column chunk: pass 1 = chunk max (wmma+fma+max),
//    pass 2 = sum exp2 against chunk max (ONE v_exp_f32 per element),
//    then one (m,s) merge per chunk.  Cross-lane merge via shfl_xor at end.
//    raw_mode=0: f_out = 0.5*(f_in + ft);  raw_mode=1: f_out = ft (F_a).
// ---------------------------------------------------------------------------
__global__ void __launch_bounds__(256) xx_kernel(const float* __restrict__ y,
                                                 float* __restrict__ ws,
                                                 int fin_off, int fout_off,
                                                 float eps, int raw_mode) {
  constexpr int CH = 1024;                 // column chunk staged in LDS
  __shared__ float4 Rp[128];               // this block's 128 row points (+|x|^2)
  __shared__ float4 Pc[CH];                // column points (+|x|^2)
  __shared__ float  Hc[CH];                // (h_j - 0.5|x_j|^2/eps) * log2e
  const int bc = blockIdx.y, b = bc / 3;
  const float inv = 1.0f / eps;
  const float s1  = inv * kLOG2E;
  const int tid = threadIdx.x;
  const int block_row0 = blockIdx.x * 128;

  if (tid < 128) {
    int i = block_row0 + tid;
    const float* yp = y + ((size_t)b * kN + i) * 6;
    float x0 = yp[0], x1 = yp[1], x2 = yp[2];
    Rp[tid] = make_float4(x0, x1, x2, x0 * x0 + x1 * x1 + x2 * x2);
  }
  __syncthreads();

  const int lane = tid & 31, wave = tid >> 5;
  const int half = lane >> 4, l16 = lane & 15;
  const int wrow0 = wave * 16;

  // A matrix (16x4 f32): lanes 0-15 -> (K0,K1)=(c0,c1); lanes 16-31 -> (K2,K3)=(c2,0)
  float4 pr = Rp[wrow0 + l16];
  v2f A; A[0] = half ? pr.z : pr.x; A[1] = half ? 0.0f : pr.y;

  float hhi[8], m[8], s[8];
#pragma unroll
  for (int r = 0; r < 8; ++r) {
    hhi[r] = -0.5f * Rp[wrow0 + r + 8 * half].w * s1;   // row constant (epilogue only)
    m[r] = -3.0e38f; s[r] = 0.0f;
  }

  for (int c0 = 0; c0 < kN; c0 += CH) {
    __syncthreads();
    for (int j = tid; j < CH; j += 256) {
      int jj = c0 + j;
      const float* yp = y + ((size_t)b * kN + jj) * 6;
      float x0 = yp[0], x1 = yp[1], x2 = yp[2];
      float xn = x0 * x0 + x1 * x1 + x2 * x2;
      Pc[j] = make_float4(x0, x1, x2, xn);
      float h = ws[OFF_ALOG + bc * kN + jj] + ws[fin_off + bc * kN + jj] * inv;
      Hc[j] = (h - 0.5f * xn * inv) * kLOG2E;
    }
    __syncthreads();

    // ---- pass 1: chunk max per D-slot (wmma + fma + max only) ----
    float mc[8];
#pragma unroll
    for (int r = 0; r < 8; ++r) mc[r] = -3.0e38f;
    for (int jt = 0; jt < CH / 16; ++jt) {
      float4 pc = Pc[jt * 16 + l16];
      v2f Bv; Bv[0] = half ? pc.z : pc.x; Bv[1] = half ? 0.0f : pc.y;
      float hc = Hc[jt * 16 + l16];
      v8f cz = {0.f, 0.f, 0.f, 0.f, 0.f, 0.f, 0.f, 0.f};
      v8f d = __builtin_amdgcn_wmma_f32_16x16x4_f32(false, A, false, Bv,
                                                    (short)0, cz, false, false);
#pragma unroll
      for (int r = 0; r < 8; ++r)
        mc[r] = fmaxf(mc[r], fmaf(d[r], s1, hc));
    }

    // ---- pass 2: sum exp2 against chunk max (one exp per element) ----
    float sc[8];
#pragma unroll
    for (int r = 0; r < 8; ++r) sc[r] = 0.0f;
    for (int jt = 0; jt < CH / 16; ++jt) {
      float4 pc = Pc[jt * 16 + l16];
      v2f Bv; Bv[0] = half ? pc.z : pc.x; Bv[1] = half ? 0.0f : pc.y;
      float hc = Hc[jt * 16 + l16];
      v8f cz = {0.f, 0.f, 0.f, 0.f, 0.f, 0.f, 0.f, 0.f};
      v8f d = __builtin_amdgcn_wmma_f32_16x16x4_f32(false, A, false, Bv,
                                                    (short)0, cz, false, false);
#pragma unroll
      for (int r = 0; r < 8; ++r) {
        float u = fmaf(d[r], s1, hc);
        sc[r] += dexp2(u - mc[r]);
      }
    }

    // ---- merge chunk (mc,sc) into running (m,s): once per chunk ----
#pragma unroll
    for (int r = 0; r < 8; ++r) {
      float M = fmaxf(m[r], mc[r]);
      s[r] = s[r] * dexp2(m[r] - M) + sc[r] * dexp2(mc[r] - M);
      m[r] = M;
    }
  }

  // merge (m,s) across the 16 column-lanes (xor<16 stays within each half)
#pragma unroll
  for (int r = 0; r < 8; ++r) {
#pragma unroll
    for (int off = 1; off < 16; off <<= 1) {
      float mo = __shfl_xor(m[r], off);
      float so = __shfl_xor(s[r], off);
      float M  = fmaxf(m[r], mo);
      s[r] = s[r] * dexp2(m[r] - M) + so * dexp2(mo - M);
      m[r] = M;
    }
  }

  if (l16 == 0) {
#pragma unroll
    for (int r = 0; r < 8; ++r) {
      int i = block_row0 + wrow0 + r + 8 * half;
      float ft = -eps * kLN2 * (m[r] + hhi[r] + dlog2(s[r]));
      float res = raw_mode ? ft : 0.5f * (ws[fin_off + bc * kN + i] + ft);
      ws[fout_off + bc * kN + i] = res;
    }
  }
}

// ---------------------------------------------------------------------------
// 7) fully fused C_yy (128x128) sym loop + final G_b: one block per (b,c).
// ---------------------------------------------------------------------------
__global__ void yy_kernel(float* __restrict__ ws) {
  __shared__ float4 pp[kK];
  __shared__ float  hv[kK];
  const int bc = blockIdx.x, t = threadIdx.x;     // 128 threads
  float4 p = ((const float4*)(ws + OFF_PTS4))[bc * kK + t];
  pp[t] = p;
  float blog = ws[OFF_BLOG + bc * kK + t];
  float gb = 0.0f;
  __syncthreads();
  for (int it = 0; it <= 12; ++it) {
    float eps = (it < 12) ? d_EPSS[it] : 0.0025f;
    float inv = 1.0f / eps;
    float s1  = inv * kLOG2E;
    hv[t] = (blog + gb * inv) * kLOG2E;           // exp2 domain
    __syncthreads();
    float mm = -3.0e38f;
    for (int j = 0; j < kK; ++j) {
      float4 q = pp[j];
      float C = 0.5f * (p.w + q.w) - (p.x * q.x + p.y * q.y + p.z * q.z);
      mm = fmaxf(mm, hv[j] - C * s1);
    }
    float ss = 0.0f;
    for (int j = 0; j < kK; ++j) {
      float4 q = pp[j];
      float C = 0.5f * (p.w + q.w) - (p.x * q.x + p.y * q.y + p.z * q.z);
      ss += dexp2(hv[j] - C * s1 - mm);
    }
    float sm = -eps * kLN2 * (mm + dlog2(ss));
    __syncthreads();
    if (it < 12) gb = 0.5f * (gb + sm);
    else ws[OFF_GXB + bc * kK + t] = sm;
    __syncthreads();
  }
}

// ---------------------------------------------------------------------------
// 8) deterministic reductions: per-(b,c) partial, then serial sum into d_out.
// ---------------------------------------------------------------------------
__global__ void reduce_kernel(const float* __restrict__ y, float* __restrict__ ws) {
  const int bc = blockIdx.x, b = bc / 3, c = bc % 3;
  float acc = 0.0f;
  for (int i = threadIdx.x; i < kN; i += 256) {
    float a = y[((size_t)b * kN + i) * 6 + 3 + c];
    acc += a * (ws[OFF_FXAB + bc * kN + i] - ws[OFF_FXA + bc * kN + i]);
  }
  for (int j = threadIdx.x; j < kK; j += 256) {
    float bv = ws[OFF_VALS + bc * kK + j];
    acc += bv * (ws[OFF_GXAB + bc * kK + j] - ws[OFF_GXB + bc * kK + j]);
  }
  __shared__ float red[256];
  red[threadIdx.x] = acc; __syncthreads();
  for (int s = 128; s > 0; s >>= 1) {
    if (threadIdx.x < s) red[threadIdx.x] += red[threadIdx.x + s];
    __syncthreads();
  }
  if (threadIdx.x == 0) ws[OFF_PART + bc] = red[0];
}

__global__ void final_kernel(const float* __restrict__ ws, float* __restrict__ out) {
  float t = 0.0f;
  for (int i = 0; i < kBC; ++i) t += ws[OFF_PART + i];
  out[0] = t;
}

// ---------------------------------------------------------------------------
extern "C" void kernel_launch(void* const* d_in, const int* in_sizes, int n_in,
                              void* d_out, int out_size, void* d_ws, size_t ws_size,
                              hipStream_t stream) {
  const float* yhat = (const float*)d_in[0];   // (B,3,V)
  const float* y    = (const float*)d_in[1];   // (B,N,6)
  const float* vox  = (const float*)d_in[2];   // (V,3)
  float* ws  = (float*)d_ws;
  float* out = (float*)d_out;

  static const float EPS_H[12] = {4.0f, 1.0f, 0.25f, 0.0625f, 0.015625f, 0.00390625f,
                                  0.0025f, 0.0025f, 0.0025f, 0.0025f, 0.0025f, 0.0025f};
  const float eps_fin = 0.0025f;

  topk_kernel<<<kBC, 256, 0, stream>>>(yhat, vox, ws);
  prep_kernel<<<dim3(kN / 256, kBC), 256, 0, stream>>>(y, ws);

  // _loop_ab over C_xy
  for (int it = 0; it < 12; ++it) {
    ftxy_kernel<<<dim3(kN / 256, kBC), 256, 0, stream>>>(y, ws, OFF_FTMP, EPS_H[it]);
    gtxy_kernel<<<dim3(kK, kBC), 256, 0, stream>>>(y, ws, OFF_GTMP, EPS_H[it]);
    upd_kernel<<<dim3(kN / 256, kBC), 256, 0, stream>>>(ws);
  }
  // F_ab, G_ab at eps = BLUR^2 using converged f0,g0
  ftxy_kernel<<<dim3(kN / 256, kBC), 256, 0, stream>>>(y, ws, OFF_FXAB, eps_fin);
  gtxy_kernel<<<dim3(kK, kBC), 256, 0, stream>>>(y, ws, OFF_GXAB, eps_fin);

  // _loop_sym over C_xx (WMMA kernel), double-buffered, then F_a
  int fin = OFF_FA0A, fo = OFF_FA0B;
  for (int it = 0; it < 12; ++it) {
    xx_kernel<<<dim3(kN / 128, kBC), 256, 0, stream>>>(y, ws, fin, fo, EPS_H[it], 0);
    int tmp = fin; fin = fo; fo = tmp;
  }
  xx_kernel<<<dim3(kN / 128, kBC), 256, 0, stream>>>(y, ws, fin, OFF_FXA, eps_fin, 1);

  // _loop_sym over C_yy + G_b, fused
  yy_kernel<<<kBC, 128, 0, stream>>>(ws);

  reduce_kernel<<<kBC, 256, 0, stream>>>(y, ws);
  final_kernel<<<1, 1, 0, stream>>>(ws, out);
}